// StyleLossDiag_13640816132828
// MI455X (gfx1250) — compile-verified
//
#include <hip/hip_runtime.h>

// StyleLossDiag on MI455X (gfx1250, wave32).
// diag(gram(x)) = per-row sum of squares of an 8192x4096 f32 matrix.
// HBM-bound: 134 MB @ 23.3 TB/s ~= 5.8 us floor. We feed the matrix unit
// with (x .* x) @ ones using V_WMMA_F32_16X16X4_F32: the 16x4 f32 A-layout
// (lanes 0-15: K=k,k+1; lanes 16-31: K=k+2,k+3) is exactly one contiguous
// float2 global load per lane, so loads -> v_mul -> v_wmma with no shuffles.

typedef float v2f __attribute__((ext_vector_type(2)));
typedef float v8f __attribute__((ext_vector_type(8)));

#define ROWS 8192            // A*B
#define COLS 4096            // C*D
#define KCHUNKS 4
#define KLEN (COLS / KCHUNKS)        // 1024 columns per wave
#define WAVES_PER_BLOCK 8
#define NORM (1.0f / 33554432.0f)    // 1/(A*B*C*D)

// Phase 1: each wave computes sum-of-squares for 16 rows over a 1024-col slice.
// 512 row-groups x 4 k-chunks = 2048 waves = 256 blocks of 256 threads.
__global__ __launch_bounds__(256) void StyleLossDiag_rowsumsq_wmma(
    const float* __restrict__ x, float* __restrict__ partial) {
  const int tid = threadIdx.x;
  const int lane = tid & 31;
  const int wid = blockIdx.x * WAVES_PER_BLOCK + (tid >> 5);  // 0..2047
  const int rowGroup = wid >> 2;   // 0..511
  const int kChunk   = wid & 3;    // 0..3
  const int baseRow  = rowGroup << 4;
  const int mrow = lane & 15;      // matrix row within the 16-row group
  const int hi   = lane >> 4;      // 0: K=k,k+1   1: K=k+2,k+3

  const float* p = x + (size_t)(baseRow + mrow) * COLS
                     + (size_t)kChunk * KLEN + (hi << 1);

  v8f acc0 = {}, acc1 = {}, acc2 = {}, acc3 = {};
  v2f ones; ones.x = 1.0f; ones.y = 1.0f;   // B = 4x16 all-ones

  // 1024 cols / (4 per WMMA) = 256 WMMA, unrolled x4 with independent
  // accumulators to break the C-operand dependency chain.
  for (int k = 0; k < KLEN; k += 16) {
    v2f d0 = *(const v2f*)(p + k);
    v2f d1 = *(const v2f*)(p + k + 4);
    v2f d2 = *(const v2f*)(p + k + 8);
    v2f d3 = *(const v2f*)(p + k + 12);
    v2f a0; a0.x = d0.x * d0.x; a0.y = d0.y * d0.y;
    v2f a1; a1.x = d1.x * d1.x; a1.y = d1.y * d1.y;
    v2f a2; a2.x = d2.x * d2.x; a2.y = d2.y * d2.y;
    v2f a3; a3.x = d3.x * d3.x; a3.y = d3.y * d3.y;
    // D = A_sq x Ones + C : every column of D holds the 16 row-sums.
    acc0 = __builtin_amdgcn_wmma_f32_16x16x4_f32(false, a0, false, ones,
                                                 (short)0, acc0, false, false);
    acc1 = __builtin_amdgcn_wmma_f32_16x16x4_f32(false, a1, false, ones,
                                                 (short)0, acc1, false, false);
    acc2 = __builtin_amdgcn_wmma_f32_16x16x4_f32(false, a2, false, ones,
                                                 (short)0, acc2, false, false);
    acc3 = __builtin_amdgcn_wmma_f32_16x16x4_f32(false, a3, false, ones,
                                                 (short)0, acc3, false, false);
  }

  v8f acc;
#pragma unroll
  for (int j = 0; j < 8; ++j) acc[j] = (acc0[j] + acc1[j]) + (acc2[j] + acc3[j]);

  // D layout: VGPR j = row j (lanes 0-15) / row j+8 (lanes 16-31), all
  // columns identical. Lane 0 stores rows 0..7, lane 16 stores rows 8..15.
  if (mrow == 0) {
    float* outp = partial + kChunk * ROWS + baseRow + (hi << 3);
#pragma unroll
    for (int j = 0; j < 8; ++j) outp[j] = acc[j];
  }
}

// Phase 2: combine the 4 k-chunk partials per row, scale, MSE vs target, mean.
__global__ __launch_bounds__(256) void StyleLossDiag_finalize(
    const float* __restrict__ partial, const float* __restrict__ target,
    float* __restrict__ out) {
  __shared__ float red[256];
  const int tid = threadIdx.x;
  float local = 0.0f;
  for (int r = tid; r < ROWS; r += 256) {
    float s = (partial[r] + partial[ROWS + r]) +
              (partial[2 * ROWS + r] + partial[3 * ROWS + r]);
    float g = s * NORM;
    float d = g - target[r];
    local += d * d;
  }
  red[tid] = local;
  __syncthreads();
  for (int off = 128; off > 0; off >>= 1) {
    if (tid < off) red[tid] += red[tid + off];
    __syncthreads();
  }
  if (tid == 0) out[0] = red[0] * (1.0f / 8192.0f);
}

extern "C" void kernel_launch(void* const* d_in, const int* in_sizes, int n_in,
                              void* d_out, int out_size, void* d_ws, size_t ws_size,
                              hipStream_t stream) {
  (void)in_sizes; (void)n_in; (void)out_size; (void)ws_size;
  const float* x      = (const float*)d_in[0];   // 16*512*64*64 f32
  const float* target = (const float*)d_in[1];   // 8192 f32
  float* out = (float*)d_out;                    // 1 f32
  float* partial = (float*)d_ws;                 // 4 * 8192 f32 = 128 KB

  StyleLossDiag_rowsumsq_wmma<<<256, 256, 0, stream>>>(x, partial);
  StyleLossDiag_finalize<<<1, 256, 0, stream>>>(partial, target, out);
}